// MaskedGCN_17162689315356
// MI455X (gfx1250) — compile-verified
//
#include <hip/hip_runtime.h>
#include <hip/hip_bf16.h>

typedef __attribute__((ext_vector_type(2))) float v2f;
typedef __attribute__((ext_vector_type(8))) float v8f;

#define NNODE 16384
#define NEDGE 262144
#define F_IN  512
#define F_HID 256
#define F_OUT 64

// ---------------------------------------------------------------------------
// degree: deg[src[e]] += 1
// ---------------------------------------------------------------------------
__global__ void degree_k(const int* __restrict__ src, float* __restrict__ deg, int E) {
    int e = blockIdx.x * blockDim.x + threadIdx.x;
    if (e < E) atomicAdd(&deg[src[e]], 1.0f);
}

// ---------------------------------------------------------------------------
// mask accumulation: m[src] += w * (x[src]-x[tgt])^2   (per feature, float4)
// ---------------------------------------------------------------------------
template <int F>
__global__ void mask_accum_k(const float* __restrict__ x, const int* __restrict__ src,
                             const int* __restrict__ tgt, const float* __restrict__ w,
                             float* __restrict__ m, int E) {
    const int per = F / 4;
    long long t = (long long)blockIdx.x * blockDim.x + threadIdx.x;
    if (t >= (long long)E * per) return;
    int e  = (int)(t / per);
    int f  = (int)(t % per) * 4;
    int s  = src[e], g = tgt[e];
    float we = w[e];
    float4 xs = *(const float4*)(x + (size_t)s * F + f);
    float4 xt = *(const float4*)(x + (size_t)g * F + f);
    float dx = xs.x - xt.x, dy = xs.y - xt.y, dz = xs.z - xt.z, dw = xs.w - xt.w;
    float* mp = m + (size_t)s * F + f;
    atomicAdd(mp + 0, we * dx * dx);
    atomicAdd(mp + 1, we * dy * dy);
    atomicAdd(mp + 2, we * dz * dz);
    atomicAdd(mp + 3, we * dw * dw);
}

// ---------------------------------------------------------------------------
// apply mask: out = exp(-(m/sigma^2)/deg) * x    (elementwise, may be in place)
// ---------------------------------------------------------------------------
template <int F>
__global__ void apply_mask_k(const float* __restrict__ x, const float* __restrict__ m,
                             const float* __restrict__ sigma, const float* __restrict__ deg,
                             float* __restrict__ out, int N) {
    const int per = F / 4;
    long long t = (long long)blockIdx.x * blockDim.x + threadIdx.x;
    if (t >= (long long)N * per) return;
    int n = (int)(t / per);
    int f = (int)(t % per) * 4;
    float dinv = 1.0f / fmaxf(deg[n], 1.0f);
    float4 xv = *(const float4*)(x + (size_t)n * F + f);
    float4 mv = *(const float4*)(m + (size_t)n * F + f);
    float4 sv = *(const float4*)(sigma + f);
    float4 o;
    o.x = __expf(-(mv.x / (sv.x * sv.x)) * dinv) * xv.x;
    o.y = __expf(-(mv.y / (sv.y * sv.y)) * dinv) * xv.y;
    o.z = __expf(-(mv.z / (sv.z * sv.z)) * dinv) * xv.z;
    o.w = __expf(-(mv.w / (sv.w * sv.w)) * dinv) * xv.w;
    *(float4*)(out + (size_t)n * F + f) = o;
}

// ---------------------------------------------------------------------------
// GEMM: Y[N,Fo] = X[N,Fi] @ W[Fo,Fi]^T + b   via V_WMMA_F32_16X16X4_F32
// One wave32 computes a 16x64 output tile: 4 independent accumulator chains,
// A fragment loaded once and reused by 4 WMMAs (4x less A traffic, 4-way ILP).
// A 16x4 layout: lanes 0-15 hold K={0,1}, lanes 16-31 hold K={2,3}.
// C/D 16x16 f32: VGPR r -> row r (lanes 0-15) / row r+8 (lanes 16-31), col=lane&15.
// ---------------------------------------------------------------------------
template <int Fi>
__global__ void gemm_wmma_k(const float* __restrict__ X, const float* __restrict__ W,
                            const float* __restrict__ bias, float* __restrict__ Y,
                            int N, int Fo) {
    int wave    = (blockIdx.x * blockDim.x + threadIdx.x) >> 5;
    int lane    = threadIdx.x & 31;
    int groupsO = Fo >> 6;               // groups of 4 O-tiles (64 output cols)
    int tM      = wave / groupsO;
    int gO      = wave - tM * groupsO;
    if (tM * 16 >= N) return;            // wave-uniform exit: EXEC stays all-1s

    int half = lane >> 4;                // 0 or 1
    int idx  = lane & 15;
    int klo  = half * 2;
    const float* xrow  = X + (size_t)(tM * 16 + idx) * Fi;       // A row (M = idx)
    const float* wrow0 = W + (size_t)(gO * 64 + idx) * Fi;       // B[k][o] = W[o][k]
    const float* wrow1 = wrow0 + (size_t)16 * Fi;
    const float* wrow2 = wrow0 + (size_t)32 * Fi;
    const float* wrow3 = wrow0 + (size_t)48 * Fi;

    v8f acc0 = {}, acc1 = {}, acc2 = {}, acc3 = {};
    for (int k = 0; k < Fi; k += 8) {
        v2f a0, a1;
        a0.x = xrow[k + klo];     a0.y = xrow[k + klo + 1];
        a1.x = xrow[k + 4 + klo]; a1.y = xrow[k + 4 + klo + 1];
        v2f b00, b01, b10, b11, b20, b21, b30, b31;
        b00.x = wrow0[k + klo];     b00.y = wrow0[k + klo + 1];
        b01.x = wrow0[k + 4 + klo]; b01.y = wrow0[k + 4 + klo + 1];
        b10.x = wrow1[k + klo];     b10.y = wrow1[k + klo + 1];
        b11.x = wrow1[k + 4 + klo]; b11.y = wrow1[k + 4 + klo + 1];
        b20.x = wrow2[k + klo];     b20.y = wrow2[k + klo + 1];
        b21.x = wrow2[k + 4 + klo]; b21.y = wrow2[k + 4 + klo + 1];
        b30.x = wrow3[k + klo];     b30.y = wrow3[k + klo + 1];
        b31.x = wrow3[k + 4 + klo]; b31.y = wrow3[k + 4 + klo + 1];
        acc0 = __builtin_amdgcn_wmma_f32_16x16x4_f32(false, a0, false, b00, (short)0, acc0, false, false);
        acc1 = __builtin_amdgcn_wmma_f32_16x16x4_f32(false, a0, false, b10, (short)0, acc1, false, false);
        acc2 = __builtin_amdgcn_wmma_f32_16x16x4_f32(false, a0, false, b20, (short)0, acc2, false, false);
        acc3 = __builtin_amdgcn_wmma_f32_16x16x4_f32(false, a0, false, b30, (short)0, acc3, false, false);
        acc0 = __builtin_amdgcn_wmma_f32_16x16x4_f32(false, a1, false, b01, (short)0, acc0, false, false);
        acc1 = __builtin_amdgcn_wmma_f32_16x16x4_f32(false, a1, false, b11, (short)0, acc1, false, false);
        acc2 = __builtin_amdgcn_wmma_f32_16x16x4_f32(false, a1, false, b21, (short)0, acc2, false, false);
        acc3 = __builtin_amdgcn_wmma_f32_16x16x4_f32(false, a1, false, b31, (short)0, acc3, false, false);
    }

    int row0 = tM * 16 + half * 8;
    int col0 = gO * 64 + idx;
    float bv0 = bias[col0];
    float bv1 = bias[col0 + 16];
    float bv2 = bias[col0 + 32];
    float bv3 = bias[col0 + 48];
#pragma unroll
    for (int r = 0; r < 8; ++r) {
        float* yr = Y + (size_t)(row0 + r) * Fo + col0;
        yr[0]  = acc0[r] + bv0;
        yr[16] = acc1[r] + bv1;
        yr[32] = acc2[r] + bv2;
        yr[48] = acc3[r] + bv3;
    }
}

// ---------------------------------------------------------------------------
// SpMM: out[src] += w * y[tgt]   (per feature, float4 gather + scalar atomics)
// ---------------------------------------------------------------------------
template <int F>
__global__ void spmm_k(const float* __restrict__ y, const int* __restrict__ src,
                       const int* __restrict__ tgt, const float* __restrict__ w,
                       float* __restrict__ out, int E) {
    const int per = F / 4;
    long long t = (long long)blockIdx.x * blockDim.x + threadIdx.x;
    if (t >= (long long)E * per) return;
    int e = (int)(t / per);
    int f = (int)(t % per) * 4;
    int s = src[e], g = tgt[e];
    float we = w[e];
    float4 yv = *(const float4*)(y + (size_t)g * F + f);
    float* op = out + (size_t)s * F + f;
    atomicAdd(op + 0, we * yv.x);
    atomicAdd(op + 1, we * yv.y);
    atomicAdd(op + 2, we * yv.z);
    atomicAdd(op + 3, we * yv.w);
}

// ---------------------------------------------------------------------------
// ReLU in place
// ---------------------------------------------------------------------------
__global__ void relu_k(float* __restrict__ x, long long n) {
    long long t = (long long)blockIdx.x * blockDim.x + threadIdx.x;
    if (t < n) x[t] = fmaxf(x[t], 0.0f);
}

// ---------------------------------------------------------------------------
// log_softmax over 64 columns; one 64-thread block per row
// ---------------------------------------------------------------------------
__global__ void logsoftmax64_k(const float* __restrict__ in, float* __restrict__ out, int N) {
    __shared__ float red[64];
    int n = blockIdx.x;
    int t = threadIdx.x;
    float v = in[(size_t)n * 64 + t];
    red[t] = v;
    __syncthreads();
#pragma unroll
    for (int s = 32; s > 0; s >>= 1) {
        if (t < s) red[t] = fmaxf(red[t], red[t + s]);
        __syncthreads();
    }
    float mx = red[0];
    __syncthreads();
    float ex = __expf(v - mx);
    red[t] = ex;
    __syncthreads();
#pragma unroll
    for (int s = 32; s > 0; s >>= 1) {
        if (t < s) red[t] += red[t + s];
        __syncthreads();
    }
    float lse = __logf(red[0]) + mx;
    out[(size_t)n * 64 + t] = v - lse;
}

// ---------------------------------------------------------------------------
// launch
// ---------------------------------------------------------------------------
extern "C" void kernel_launch(void* const* d_in, const int* in_sizes, int n_in,
                              void* d_out, int out_size, void* d_ws, size_t ws_size,
                              hipStream_t stream) {
    const float* feat   = (const float*)d_in[0];
    const int*   esrc   = (const int*)d_in[1];
    const int*   etgt   = (const int*)d_in[2];
    const float* ew     = (const float*)d_in[3];
    const float* W1     = (const float*)d_in[4];
    const float* b1     = (const float*)d_in[5];
    const float* sigma1 = (const float*)d_in[6];
    const float* W2     = (const float*)d_in[7];
    const float* b2     = (const float*)d_in[8];
    const float* sigma2 = (const float*)d_in[9];
    float* out = (float*)d_out;

    const int N = NNODE, E = NEDGE;

    // workspace layout (floats), aliased across phases
    float* deg = (float*)d_ws;                        // N
    float* A   = deg + N;                             // N*512 : m1 -> xm1 ; later y2|h2
    float* B   = A + (size_t)N * F_IN;                // N*256 : y1 -> m2 -> xm2
    float* C   = B + (size_t)N * F_HID;               // N*256 : h1 (relu'd)
    float* y2  = A;                                   // N*64
    float* h2  = A + (size_t)N * F_OUT;               // N*64

    const int BLK = 256;
    auto blocks = [](long long work, int blk) { return (int)((work + blk - 1) / blk); };

    // ---- degree ----
    hipMemsetAsync(deg, 0, (size_t)N * 4, stream);
    degree_k<<<blocks(E, BLK), BLK, 0, stream>>>(esrc, deg, E);

    // ---- layer 1: mask ----
    hipMemsetAsync(A, 0, (size_t)N * F_IN * 4, stream);
    mask_accum_k<F_IN><<<blocks((long long)E * (F_IN / 4), BLK), BLK, 0, stream>>>(
        feat, esrc, etgt, ew, A, E);
    apply_mask_k<F_IN><<<blocks((long long)N * (F_IN / 4), BLK), BLK, 0, stream>>>(
        feat, A, sigma1, deg, A, N);   // in-place: A = xm1

    // ---- layer 1: GEMM (WMMA f32, 16x64 tile per wave) ----
    {
        long long waves = (long long)(N / 16) * (F_HID / 64);
        gemm_wmma_k<F_IN><<<blocks(waves * 32, BLK), BLK, 0, stream>>>(A, W1, b1, B, N, F_HID);
    }

    // ---- layer 1: SpMM + ReLU ----
    hipMemsetAsync(C, 0, (size_t)N * F_HID * 4, stream);
    spmm_k<F_HID><<<blocks((long long)E * (F_HID / 4), BLK), BLK, 0, stream>>>(
        B, esrc, etgt, ew, C, E);
    relu_k<<<blocks((long long)N * F_HID, BLK), BLK, 0, stream>>>(C, (long long)N * F_HID);

    // ---- layer 2: mask ----
    hipMemsetAsync(B, 0, (size_t)N * F_HID * 4, stream);   // reuse y1 space for m2
    mask_accum_k<F_HID><<<blocks((long long)E * (F_HID / 4), BLK), BLK, 0, stream>>>(
        C, esrc, etgt, ew, B, E);
    apply_mask_k<F_HID><<<blocks((long long)N * (F_HID / 4), BLK), BLK, 0, stream>>>(
        C, B, sigma2, deg, B, N);      // in-place: B = xm2

    // ---- layer 2: GEMM (WMMA f32, 16x64 tile per wave) ----
    {
        long long waves = (long long)(N / 16) * (F_OUT / 64);
        gemm_wmma_k<F_HID><<<blocks(waves * 32, BLK), BLK, 0, stream>>>(B, W2, b2, y2, N, F_OUT);
    }

    // ---- layer 2: SpMM ----
    hipMemsetAsync(h2, 0, (size_t)N * F_OUT * 4, stream);
    spmm_k<F_OUT><<<blocks((long long)E * (F_OUT / 4), BLK), BLK, 0, stream>>>(
        y2, esrc, etgt, ew, h2, E);

    // ---- log_softmax ----
    logsoftmax64_k<<<N, 64, 0, stream>>>(h2, out, N);
}